// ModularGNN_90649579749762
// MI455X (gfx1250) — compile-verified
//
#include <hip/hip_runtime.h>

typedef float v2f __attribute__((ext_vector_type(2)));
typedef float v8f __attribute__((ext_vector_type(8)));

#define NGRAPHS 16

// ---------------------------------------------------------------------------
// H[N,Dout] = X[N,K] @ W[K,Dout] + bias   (fp32-exact, V_WMMA_F32_16X16X4_F32)
// One wave32 computes one 16x16 output tile; K advances 4 per WMMA.
// A frag (16x4):  lanes 0-15 -> M=lane, K={k0,k0+1}; lanes 16-31 -> K={k0+2,k0+3}
// B frag (4x16):  lanes 0-15 -> N=lane, K={k0,k0+1}; lanes 16-31 -> K={k0+2,k0+3}
// C/D (16x16 f32, 8 VGPRs): VGPR v, lanes 0-15 -> M=v; lanes 16-31 -> M=v+8
// ---------------------------------------------------------------------------
__global__ void __launch_bounds__(256)
gemm_bias_wmma(const float* __restrict__ X, const float* __restrict__ W,
               const float* __restrict__ bias, float* __restrict__ H,
               int N, int K, int Dout) {
    const int lane    = threadIdx.x & 31;
    const int ntN     = Dout >> 4;                       // 16-wide col tiles
    const int wave    = blockIdx.x * (blockDim.x >> 5) + (threadIdx.x >> 5);
    const int totalWv = (N >> 4) * ntN;
    if (wave >= totalWv) return;                         // wave-uniform guard

    const int tM      = wave / ntN;
    const int tN      = wave % ntN;
    const int halfSel = lane >> 4;                       // 0 or 1
    const int l15     = lane & 15;
    const int kOff    = halfSel * 2;                     // K sub-offset 0 / 2
    const long long rowA = (long long)tM * 16 + l15;     // A row (M)
    const int colB    = tN * 16 + l15;                   // B col (N)

    const float* __restrict__ xrow = X + rowA * K;

    v8f acc = {0.f, 0.f, 0.f, 0.f, 0.f, 0.f, 0.f, 0.f};

    for (int k0 = 0; k0 < K; k0 += 4) {
        v2f a, b;
        a.x = xrow[k0 + kOff];
        a.y = xrow[k0 + kOff + 1];
        b.x = W[(long long)(k0 + kOff)     * Dout + colB];
        b.y = W[(long long)(k0 + kOff + 1) * Dout + colB];
        acc = __builtin_amdgcn_wmma_f32_16x16x4_f32(
            /*neg_a=*/false, a, /*neg_b=*/false, b,
            /*c_mod=*/(short)0, acc, /*reuse_a=*/false, /*reuse_b=*/false);
    }

    const float bv    = bias[colB];
    const long long r0 = (long long)tM * 16 + halfSel * 8;
#pragma unroll
    for (int v = 0; v < 8; ++v) {
        H[(r0 + v) * Dout + colB] = acc[v] + bv;
    }
}

// ---------------------------------------------------------------------------
// out[dst[e], :] += h[src[e], :] * w[e]
// 1 thread per (edge, 4-channel chunk): float4 gather (coalesced, L2-resident)
// + 4 scalar global_atomic_add_f32.
// ---------------------------------------------------------------------------
__global__ void __launch_bounds__(256)
scatter_edges(const float* __restrict__ h, const int* __restrict__ src,
              const int* __restrict__ dst, const float* __restrict__ ea,
              float* __restrict__ out, int E, int D) {
    const int chunks = D >> 2;
    long long id = (long long)blockIdx.x * blockDim.x + threadIdx.x;
    if (id >= (long long)E * chunks) return;
    const int e = (int)(id / chunks);
    const int c = (int)(id % chunks);

    const int s = src[e];
    const int d = dst[e];
    const float w = ea[e];

    const float4 m = *(const float4*)(h + (long long)s * D + c * 4);
    float* op = out + (long long)d * D + c * 4;
    atomicAdd(op + 0, m.x * w);
    atomicAdd(op + 1, m.y * w);
    atomicAdd(op + 2, m.z * w);
    atomicAdd(op + 3, m.w * w);
}

__global__ void zero_f32(float* __restrict__ p, int n) {
    int i = blockIdx.x * blockDim.x + threadIdx.x;
    if (i < n) p[i] = 0.f;
}

__global__ void __launch_bounds__(256)
pool_acc(const float* __restrict__ h, const int* __restrict__ batch,
         float* __restrict__ sums, float* __restrict__ counts, int N, int D) {
    long long id = (long long)blockIdx.x * blockDim.x + threadIdx.x;
    if (id >= (long long)N * D) return;
    const int n = (int)(id / D);
    const int c = (int)(id % D);
    const int g = batch[n];
    atomicAdd(&sums[g * D + c], h[(long long)n * D + c]);
    if (c == 0) atomicAdd(&counts[g], 1.0f);
}

__global__ void pool_fin(const float* __restrict__ sums,
                         const float* __restrict__ counts,
                         float* __restrict__ out, int n, int D) {
    int i = blockIdx.x * blockDim.x + threadIdx.x;
    if (i < n) out[i] = sums[i] / fmaxf(counts[i / D], 1.0f);
}

// ---------------------------------------------------------------------------
extern "C" void kernel_launch(void* const* d_in, const int* in_sizes, int n_in,
                              void* d_out, int out_size, void* d_ws, size_t ws_size,
                              hipStream_t stream) {
    const float* x     = (const float*)d_in[0];
    const int*   ei    = (const int*)  d_in[1];   // [2,E] int32 (x64 disabled)
    const float* ea    = (const float*)d_in[2];
    const int*   batch = (const int*)  d_in[3];
    const float* W1 = (const float*)d_in[4]; const float* b1 = (const float*)d_in[5];
    const float* W2 = (const float*)d_in[6]; const float* b2 = (const float*)d_in[7];
    const float* W3 = (const float*)d_in[8]; const float* b3 = (const float*)d_in[9];
    float* out = (float*)d_out;

    const int E = in_sizes[2];
    const int N = in_sizes[3];
    const int DIN = 128, DH = 128, DOUT = 64;

    const int* src = ei;
    const int* dst = ei + E;

    char* ws = (char*)d_ws;
    const size_t bufBytes = (size_t)N * DIN * sizeof(float);
    float* bufA   = (float*)ws;                   // h  (GEMM output, gather src)
    float* bufB   = (float*)(ws + bufBytes);      // out (= h + scatter-sum)
    float* sums   = (float*)(ws + 2 * bufBytes);  // [16*64]
    float* counts = sums + NGRAPHS * DOUT;        // [16]

    auto launch_gemm = [&](const float* X, const float* W, const float* B,
                           float* H, int K, int Dout) {
        int waves = (N >> 4) * (Dout >> 4);
        int blocks = (waves + 7) / 8;             // 8 waves (256 thr) per block
        gemm_bias_wmma<<<blocks, 256, 0, stream>>>(X, W, B, H, N, K, Dout);
    };
    auto launch_scatter = [&](const float* h, float* o, int D) {
        long long work = (long long)E * (D >> 2);
        int blocks = (int)((work + 255) / 256);
        scatter_edges<<<blocks, 256, 0, stream>>>(h, src, dst, ea, o, E, D);
    };

    // ---- layer 1:  h1 = x@W1+b1 ; out1 = h1 + scatter(h1) ----
    launch_gemm(x, W1, b1, bufA, DIN, DH);
    hipMemcpyAsync(bufB, bufA, (size_t)N * DH * sizeof(float),
                   hipMemcpyDeviceToDevice, stream);
    launch_scatter(bufA, bufB, DH);

    // ---- layer 2 ----
    launch_gemm(bufB, W2, b2, bufA, DH, DH);
    hipMemcpyAsync(bufB, bufA, (size_t)N * DH * sizeof(float),
                   hipMemcpyDeviceToDevice, stream);
    launch_scatter(bufA, bufB, DH);

    // ---- layer 3 (Dout = 64) ----
    launch_gemm(bufB, W3, b3, bufA, DH, DOUT);
    hipMemcpyAsync(bufB, bufA, (size_t)N * DOUT * sizeof(float),
                   hipMemcpyDeviceToDevice, stream);
    launch_scatter(bufA, bufB, DOUT);

    // ---- per-graph mean pool ----
    zero_f32<<<(NGRAPHS * DOUT + NGRAPHS + 255) / 256, 256, 0, stream>>>(
        sums, NGRAPHS * DOUT + NGRAPHS);
    {
        long long work = (long long)N * DOUT;
        pool_acc<<<(int)((work + 255) / 256), 256, 0, stream>>>(
            bufB, batch, sums, counts, N, DOUT);
    }
    pool_fin<<<(NGRAPHS * DOUT + 255) / 256, 256, 0, stream>>>(
        sums, counts, out, NGRAPHS * DOUT, DOUT);
}